// GIN_50577534878114
// MI455X (gfx1250) — compile-verified
//
#include <hip/hip_runtime.h>
#include <hip/hip_bf16.h>

typedef float v2f __attribute__((ext_vector_type(2)));
typedef float v8f __attribute__((ext_vector_type(8)));

#define GIN_D 64

// ---------------------------------------------------------------------------
// Scatter-add aggregation: agg[dst[e]] += h[src[e]] over all edges.
// 16 lanes per edge, each lane moves a float4 (row read = 256B, coalesced).
// Atomics stay in L2 (working set < 192MB), hardware f32 add via unsafeAtomicAdd.
// ---------------------------------------------------------------------------
__global__ void __launch_bounds__(256) gin_scatter(
    const float* __restrict__ h, const long long* __restrict__ src,
    const long long* __restrict__ dst, float* __restrict__ agg, int nE) {
  int gid = blockIdx.x * 256 + threadIdx.x;
  int e = gid >> 4;
  if (e >= nE) return;
  int c = (gid & 15) << 2;                    // float4 channel base
  long long s = src[e];
  long long d = dst[e];
  const float4 v = *(const float4*)(h + (size_t)s * GIN_D + c);
  float* p = agg + (size_t)d * GIN_D + c;
  unsafeAtomicAdd(p + 0, v.x);
  unsafeAtomicAdd(p + 1, v.y);
  unsafeAtomicAdd(p + 2, v.z);
  unsafeAtomicAdd(p + 3, v.w);
}

// ---------------------------------------------------------------------------
// Layer 1: h1 = relu((agg + (1+eps0)*x) @ W0^T)
// One wave32 computes a 16(row) x 64(col) tile with V_WMMA_F32_16X16X4_F32,
// K=64 in 16 chunks of 4. W0 staged in LDS once per block (8 waves).
// Fragment layouts per CDNA5 ISA 7.12.2:
//   A 16x4 f32 : lane%16 = M, (lane/16, vgpr) = K in {0,1,2,3}
//   B 4x16 f32 : lane%16 = N, (lane/16, vgpr) = K  (B[k][n] = W[n][k])
//   C/D 16x16  : vgpr j -> M = j + 8*(lane/16), lane%16 = N
// ---------------------------------------------------------------------------
__global__ void __launch_bounds__(256) gin_layer1(
    const float* __restrict__ x, const float* __restrict__ agg,
    const float* __restrict__ W0, const float* __restrict__ eps0,
    float* __restrict__ h1, int N, int nTiles) {
  __shared__ float sW[GIN_D * GIN_D];
  for (int i = threadIdx.x; i < GIN_D * GIN_D; i += 256) sW[i] = W0[i];
  __syncthreads();

  int wave = threadIdx.x >> 5;
  int lane = threadIdx.x & 31;
  int tile = blockIdx.x * 8 + wave;
  if (tile >= nTiles) return;

  float c0 = 1.0f + eps0[0];
  int lm = lane & 15;                 // M (for A) / N (for B)
  int kh = (lane >> 4) << 1;          // K offset from lane half: 0 or 2
  int m = tile * 16 + lm;
  if (m >= N) m = N - 1;              // clamp loads; stores are guarded

  v8f acc[4] = {};
  for (int kc = 0; kc < 16; ++kc) {
    int k = kc * 4 + kh;
    v2f xa = *(const v2f*)(x + (size_t)m * GIN_D + k);
    v2f ga = *(const v2f*)(agg + (size_t)m * GIN_D + k);
    v2f a;
    a.x = ga.x + c0 * xa.x;
    a.y = ga.y + c0 * xa.y;
#pragma unroll
    for (int nt = 0; nt < 4; ++nt) {
      int n = nt * 16 + lm;
      v2f b = *(const v2f*)(sW + n * GIN_D + k);
      acc[nt] = __builtin_amdgcn_wmma_f32_16x16x4_f32(
          false, a, false, b, (short)0, acc[nt], false, false);
    }
  }

  int mb = tile * 16 + ((lane >> 4) << 3);
#pragma unroll
  for (int nt = 0; nt < 4; ++nt) {
#pragma unroll
    for (int j = 0; j < 8; ++j) {
      int row = mb + j;
      if (row < N) {
        float v = acc[nt][j];
        h1[(size_t)row * GIN_D + nt * 16 + lm] = v > 0.0f ? v : 0.0f;
      }
    }
  }
}

// ---------------------------------------------------------------------------
// Layer 2 fused with output head:
//   h2  = relu((agg1 + (1+eps1)*h1) @ W1^T)
//   out = (h1 @ Wm^T + h2) * 0.5
// Two WMMA chains per wave (W1 on fused input, Wm on raw h1); h2 never hits
// memory. W1 and Wm both staged in LDS (32KB of the 320KB/WGP).
// ---------------------------------------------------------------------------
__global__ void __launch_bounds__(256) gin_layer2(
    const float* __restrict__ h1, const float* __restrict__ agg,
    const float* __restrict__ W1, const float* __restrict__ Wm,
    const float* __restrict__ eps1, float* __restrict__ out,
    int N, int nTiles) {
  __shared__ float sW1[GIN_D * GIN_D];
  __shared__ float sWm[GIN_D * GIN_D];
  for (int i = threadIdx.x; i < GIN_D * GIN_D; i += 256) {
    sW1[i] = W1[i];
    sWm[i] = Wm[i];
  }
  __syncthreads();

  int wave = threadIdx.x >> 5;
  int lane = threadIdx.x & 31;
  int tile = blockIdx.x * 8 + wave;
  if (tile >= nTiles) return;

  float c1 = 1.0f + eps1[0];
  int lm = lane & 15;
  int kh = (lane >> 4) << 1;
  int m = tile * 16 + lm;
  if (m >= N) m = N - 1;

  v8f accA[4] = {};  // W1 @ (agg + c1*h1)
  v8f accB[4] = {};  // Wm @ h1
  for (int kc = 0; kc < 16; ++kc) {
    int k = kc * 4 + kh;
    v2f hh = *(const v2f*)(h1 + (size_t)m * GIN_D + k);
    v2f gg = *(const v2f*)(agg + (size_t)m * GIN_D + k);
    v2f a2;
    a2.x = gg.x + c1 * hh.x;
    a2.y = gg.y + c1 * hh.y;
#pragma unroll
    for (int nt = 0; nt < 4; ++nt) {
      int n = nt * 16 + lm;
      v2f b1 = *(const v2f*)(sW1 + n * GIN_D + k);
      v2f bm = *(const v2f*)(sWm + n * GIN_D + k);
      accA[nt] = __builtin_amdgcn_wmma_f32_16x16x4_f32(
          false, a2, false, b1, (short)0, accA[nt], false, false);
      accB[nt] = __builtin_amdgcn_wmma_f32_16x16x4_f32(
          false, hh, false, bm, (short)0, accB[nt], false, false);
    }
  }

  int mb = tile * 16 + ((lane >> 4) << 3);
#pragma unroll
  for (int nt = 0; nt < 4; ++nt) {
#pragma unroll
    for (int j = 0; j < 8; ++j) {
      int row = mb + j;
      if (row < N) {
        float h2v = accA[nt][j];
        h2v = h2v > 0.0f ? h2v : 0.0f;
        out[(size_t)row * GIN_D + nt * 16 + lm] = 0.5f * (accB[nt][j] + h2v);
      }
    }
  }
}

extern "C" void kernel_launch(void* const* d_in, const int* in_sizes, int n_in,
                              void* d_out, int out_size, void* d_ws, size_t ws_size,
                              hipStream_t stream) {
  const float* x        = (const float*)d_in[0];
  const long long* ei   = (const long long*)d_in[1];
  const float* W0       = (const float*)d_in[2];
  const float* eps0     = (const float*)d_in[3];
  const float* W1       = (const float*)d_in[4];
  const float* eps1     = (const float*)d_in[5];
  const float* Wm       = (const float*)d_in[6];
  float* out            = (float*)d_out;

  const int N  = in_sizes[0] / GIN_D;
  const int nE = in_sizes[1] / 2;
  const long long* src = ei;
  const long long* dst = ei + nE;

  float* agg = (float*)d_ws;                       // N*D floats
  float* h1  = agg + (size_t)N * GIN_D;            // N*D floats
  const size_t aggBytes = (size_t)N * GIN_D * sizeof(float);

  const int nTiles = (N + 15) / 16;
  dim3 blk(256);
  dim3 gridS((unsigned)(((size_t)nE * 16 + 255) / 256));
  dim3 gridG((unsigned)((nTiles + 7) / 8));

  // Layer 1
  hipMemsetAsync(agg, 0, aggBytes, stream);
  gin_scatter<<<gridS, blk, 0, stream>>>(x, src, dst, agg, nE);
  gin_layer1<<<gridG, blk, 0, stream>>>(x, agg, W0, eps0, h1, N, nTiles);

  // Layer 2 + output head
  hipMemsetAsync(agg, 0, aggBytes, stream);
  gin_scatter<<<gridS, blk, 0, stream>>>(h1, src, dst, agg, nE);
  gin_layer2<<<gridG, blk, 0, stream>>>(h1, agg, W1, Wm, eps1, out, N, nTiles);
}